// FIAttention_31284541784482
// MI455X (gfx1250) — compile-verified
//
#include <hip/hip_runtime.h>
#include <math.h>

typedef __attribute__((ext_vector_type(2))) float v2f_t;
typedef __attribute__((ext_vector_type(8))) float v8f_t;

#define B_   32
#define DIM_ 1024
#define H_   8
#define DH_  128
#define KV_  4096

// ---------------------------------------------------------------------------
// Kernel A: fused Q/K/V projection  (x[32,1024] @ W[1024,1024]) using
// V_WMMA_F32_16X16X4_F32. One wave computes one 16x16 output tile.
// blockDim = 256 (8 waves); grid = (8 n-tile groups, 2 m-tiles, 3 matrices).
// ---------------------------------------------------------------------------
__global__ __launch_bounds__(256) void qkv_proj(
    const float* __restrict__ x,  const float* __restrict__ Wq,
    const float* __restrict__ Wk, const float* __restrict__ Wv,
    float* __restrict__ qo, float* __restrict__ ko, float* __restrict__ vo)
{
    const int wave  = threadIdx.x >> 5;
    const int lane  = threadIdx.x & 31;
    const int half  = lane >> 4;          // 0: lanes 0-15, 1: lanes 16-31
    const int l16   = lane & 15;
    const int nt    = blockIdx.x * 8 + wave;   // 0..63 (N tile)
    const int mt    = blockIdx.y;              // 0..1  (M tile)
    const int which = blockIdx.z;              // 0:Wq 1:Wk 2:Wv

    const float* W   = (which == 0) ? Wq : (which == 1) ? Wk : Wv;
    float*       out = (which == 0) ? qo : (which == 1) ? ko : vo;

    const int m0 = mt * 16;
    const int n0 = nt * 16;

    // A fragment base: row (m0+l16), columns k + 2*half .. +1  (contiguous b64)
    const float* ap = x + (size_t)(m0 + l16) * DIM_ + half * 2;
    // B fragment base: rows k + 2*half, k + 2*half + 1, column (n0+l16)
    const float* bp = W + (size_t)(half * 2) * DIM_ + n0 + l16;

    v8f_t acc = {};
    for (int k = 0; k < DIM_; k += 4) {
        v2f_t a, b;
        a.x = ap[k];
        a.y = ap[k + 1];
        b.x = bp[(size_t)k * DIM_];
        b.y = bp[(size_t)k * DIM_ + DIM_];
        // 8 args: (neg_a, A, neg_b, B, c_mod, C, reuse_a, reuse_b)
        acc = __builtin_amdgcn_wmma_f32_16x16x4_f32(
            false, a, false, b, (short)0, acc, false, false);
    }

    // C/D layout: VGPR r, lanes 0-15 -> M=r, lanes 16-31 -> M=r+8; N=l16
    for (int r = 0; r < 8; ++r) {
        const int m = m0 + r + 8 * half;
        out[(size_t)m * DIM_ + n0 + l16] = acc[r];
    }
}

// ---------------------------------------------------------------------------
// Kernel B: bandwidth-bound decode attention with online softmax.
// One workgroup (8 waves) per (b,h). Each lane owns 4 head dims -> one wave
// streams a full 128-float K row and V row per step as b128 loads.
// ---------------------------------------------------------------------------
__global__ __launch_bounds__(256) void decode_attn(
    const float* __restrict__ cache_k, const float* __restrict__ cache_v,
    const float* __restrict__ qbuf,    const float* __restrict__ kapp,
    const float* __restrict__ vapp,    float* __restrict__ out)
{
    __shared__ float s_m[8];
    __shared__ float s_l[8];
    __shared__ float s_acc[8][DH_];

    const int b    = blockIdx.x;
    const int h    = blockIdx.y;
    const int tid  = threadIdx.x;
    const int wave = tid >> 5;
    const int lane = tid & 31;

    const float scale = 0.08838834764831845f;   // 1/sqrt(128)
    const size_t bh   = ((size_t)b * H_ + h) * DH_;
    const size_t srow = (size_t)H_ * DH_;       // 1024 floats between s rows

    const float4 qf = *(const float4*)(qbuf + bh + 4 * lane);

    const float* kp = cache_k + (((size_t)b * KV_) * H_ + h) * DH_ + 4 * lane;
    const float* vp = cache_v + (((size_t)b * KV_) * H_ + h) * DH_ + 4 * lane;

    float m = -INFINITY, l = 0.f;
    float ax = 0.f, ay = 0.f, az = 0.f, aw = 0.f;

    for (int s = wave; s < KV_; s += 8) {
        const float* kr = kp + (size_t)s * srow;
        const float* vr = vp + (size_t)s * srow;
        // gfx1250: global_prefetch_b8 for the next row this wave will touch
        __builtin_prefetch(kr + 8 * srow, 0, 1);
        __builtin_prefetch(vr + 8 * srow, 0, 1);

        const float4 k4 = *(const float4*)kr;
        const float4 v4 = *(const float4*)vr;

        float d = qf.x * k4.x + qf.y * k4.y + qf.z * k4.z + qf.w * k4.w;
        d += __shfl_xor(d, 16, 32);
        d += __shfl_xor(d, 8, 32);
        d += __shfl_xor(d, 4, 32);
        d += __shfl_xor(d, 2, 32);
        d += __shfl_xor(d, 1, 32);

        const float logit = d * scale;
        const float mn    = fmaxf(m, logit);
        const float corr  = __expf(m - mn);
        const float p     = __expf(logit - mn);
        l  = l  * corr + p;
        ax = ax * corr + p * v4.x;
        ay = ay * corr + p * v4.y;
        az = az * corr + p * v4.z;
        aw = aw * corr + p * v4.w;
        m  = mn;
    }

    // Appended token (s = KV) handled by wave 0 only.
    if (wave == 0) {
        const float4 k4 = *(const float4*)(kapp + bh + 4 * lane);
        const float4 v4 = *(const float4*)(vapp + bh + 4 * lane);

        float d = qf.x * k4.x + qf.y * k4.y + qf.z * k4.z + qf.w * k4.w;
        d += __shfl_xor(d, 16, 32);
        d += __shfl_xor(d, 8, 32);
        d += __shfl_xor(d, 4, 32);
        d += __shfl_xor(d, 2, 32);
        d += __shfl_xor(d, 1, 32);

        const float logit = d * scale;
        const float mn    = fmaxf(m, logit);
        const float corr  = __expf(m - mn);
        const float p     = __expf(logit - mn);
        l  = l  * corr + p;
        ax = ax * corr + p * v4.x;
        ay = ay * corr + p * v4.y;
        az = az * corr + p * v4.z;
        aw = aw * corr + p * v4.w;
        m  = mn;
    }

    s_acc[wave][4 * lane + 0] = ax;
    s_acc[wave][4 * lane + 1] = ay;
    s_acc[wave][4 * lane + 2] = az;
    s_acc[wave][4 * lane + 3] = aw;
    if (lane == 0) { s_m[wave] = m; s_l[wave] = l; }
    __syncthreads();

    // Merge the 8 per-wave online-softmax partials.
    if (tid < DH_) {
        float M = s_m[0];
        for (int w = 1; w < 8; ++w) M = fmaxf(M, s_m[w]);
        float L = 0.f, O = 0.f;
        for (int w = 0; w < 8; ++w) {
            const float c = __expf(s_m[w] - M);
            L += s_l[w] * c;
            O += s_acc[w][tid] * c;
        }
        out[bh + tid] = O / L;
    }
}

// ---------------------------------------------------------------------------
extern "C" void kernel_launch(void* const* d_in, const int* in_sizes, int n_in,
                              void* d_out, int out_size, void* d_ws, size_t ws_size,
                              hipStream_t stream)
{
    const float* x  = (const float*)d_in[0];
    const float* ck = (const float*)d_in[1];
    const float* cv = (const float*)d_in[2];
    const float* Wq = (const float*)d_in[3];
    const float* Wk = (const float*)d_in[4];
    const float* Wv = (const float*)d_in[5];

    float* qo = (float*)d_ws;                 // [32][1024]
    float* ko = qo + (size_t)B_ * DIM_;       // [32][1024]
    float* vo = ko + (size_t)B_ * DIM_;       // [32][1024]
    float* o  = (float*)d_out;                // [32][8][128]

    dim3 gA(8, 2, 3);
    qkv_proj<<<gA, 256, 0, stream>>>(x, Wq, Wk, Wv, qo, ko, vo);

    dim3 gB(B_, H_);
    decode_attn<<<gB, 256, 0, stream>>>(ck, cv, qo, ko, vo, o);
}